// VQVAE_19232863551965
// MI455X (gfx1250) — compile-verified
//
#include <hip/hip_runtime.h>
#include <hip/hip_bf16.h>
#include <math.h>

typedef __attribute__((ext_vector_type(2))) float v2f;
typedef __attribute__((ext_vector_type(8))) float v8f;
typedef unsigned int v4u __attribute__((ext_vector_type(4)));
typedef int v8i __attribute__((ext_vector_type(8)));
typedef int v4i __attribute__((ext_vector_type(4)));

// ---------------- workspace layout (float offsets) ----------------
#define Z1_OFF   0u           // 64*16*128*128 = 16,777,216 floats (reused for y2)
#define Z2_OFF   16777216u    // 64*4*64*64    =  1,048,576 floats
#define D1_OFF   17825792u    // 64*4*64*64    =  1,048,576 floats
#define ST_OFF   18874368u    // stats block

// stats sub-offsets
#define ST_BN1_SUM 0     // [16] sum, [16] sumsq at +16
#define ST_BN2_SUM 32    // [4] sum,  [4] sumsq at +4
#define ST_BN3_SUM 40    // [16] sum, [16] sumsq at +16
#define ST_BN1_SC  72    // [16] scale, [16] shift
#define ST_BN2_SC  104   // [4] scale, [4] shift
#define ST_BN3_SC  112   // [16] scale, [16] shift
#define ST_PROJ    144   // [3][4]
#define ST_LOSS    156
#define ST_TOTAL   160

// ---------------- TDM: DMA a 2D f32 tile (dim1 rows x dim0 cols) global -> LDS ----------------
// Descriptor per CDNA5 ISA §8.3/§8.4. Groups 2/3 zero (tile_dim2 == 0, <=2D tensor).
__device__ __forceinline__ void tdm_load_2d_f32_to_lds(const void* gptr, unsigned lds_byte_addr,
                                                       unsigned dim0, unsigned dim1,
                                                       unsigned stride0) {
    unsigned long long ga = (unsigned long long)gptr;
    v4u g0;
    g0[0] = 1u;                                                  // count=1, user mode
    g0[1] = lds_byte_addr;                                       // lds_addr -> bits [63:32]
    g0[2] = (unsigned)ga;                                        // global_addr[31:0]
    g0[3] = (unsigned)((ga >> 32) & 0x01FFFFFFu) | (2u << 30);   // global_addr[56:32] | type=2
    v8i g1;
    g1[0] = (int)(2u << 16);          // data_size=2 (4 bytes); workgroup_mask=0 (not in cluster)
    g1[1] = (int)(dim0 << 16);        // tensor_dim0[15:0]  -> bits [63:48]
    g1[2] = (int)(dim1 << 16);        // tensor_dim1[15:0]  -> bits [95:80]
    g1[3] = (int)(dim0 << 16);        // tile_dim0          -> bits [127:112]
    g1[4] = (int)dim1;                // tile_dim1 (tile_dim2 = 0 -> 2D)
    g1[5] = (int)stride0;             // tensor_dim0_stride[31:0]
    g1[6] = 0;
    g1[7] = 0;
    v4i z4 = {0, 0, 0, 0};
#if __clang_major__ >= 23
    v8i z8 = {0, 0, 0, 0, 0, 0, 0, 0};
    __builtin_amdgcn_tensor_load_to_lds(g0, g1, z4, z4, z8, 0);
#else
    __builtin_amdgcn_tensor_load_to_lds(g0, g1, z4, z4, 0);
#endif
}

// ---------------- tiny utility kernels ----------------
__global__ void k_zero_stats(float* stats) {
    if (threadIdx.x < ST_TOTAL) stats[threadIdx.x] = 0.0f;
}

// proj[k][co] = emb[k] . dec_w1[co] + dec_b1[co]   (dec1 applied to each codebook entry)
__global__ void k_proj(const float* __restrict__ emb, const float* __restrict__ w,
                       const float* __restrict__ b, float* stats) {
    int t = threadIdx.x;
    if (t < 12) {
        int k = t >> 2, co = t & 3;
        float acc = b[co];
        for (int c = 0; c < 64; ++c) acc += emb[k * 64 + c] * w[co * 64 + c];
        stats[ST_PROJ + t] = acc;
    }
}

// scale = g*rsqrt(var+eps); shift = b - mean*scale
__global__ void k_finalize_bn(float* stats, const float* __restrict__ g,
                              const float* __restrict__ b, int sum_off, int out_off,
                              int C, float invN) {
    int c = threadIdx.x;
    if (c < C) {
        float m = stats[sum_off + c] * invN;
        float v = stats[sum_off + C + c] * invN - m * m;
        float sc = g[c] * rsqrtf(v + 1e-5f);
        stats[out_off + c] = sc;
        stats[out_off + C + c] = b[c] - m * sc;
    }
}

// ---------------- conv1: 1->16, k4 s2 p1, x(64,1,256,256) -> z1(64,16,128,128) + bn1 stats
__global__ void k1_conv1(const float* __restrict__ x, const float* __restrict__ w,
                         const float* __restrict__ bias, float* __restrict__ z1,
                         float* stats) {
    __shared__ float s_w[256];
    __shared__ float s_sum[16], s_sq[16];
    int t = threadIdx.x;
    s_w[t] = w[t];                       // 16*1*4*4 = 256, blockDim == 256
    if (t < 16) { s_sum[t] = 0.f; s_sq[t] = 0.f; }
    __syncthreads();

    int gid = blockIdx.x * 256 + t;      // over 64*128*128
    int ox = gid & 127, oy = (gid >> 7) & 127, b = gid >> 14;
    const float* xb = x + b * 65536;

    float patch[16];
#pragma unroll
    for (int ky = 0; ky < 4; ++ky) {
        int iy = 2 * oy - 1 + ky;
#pragma unroll
        for (int kx = 0; kx < 4; ++kx) {
            int ix = 2 * ox - 1 + kx;
            float v = 0.f;
            if ((unsigned)iy < 256u && (unsigned)ix < 256u) v = xb[iy * 256 + ix];
            patch[ky * 4 + kx] = v;
        }
    }
#pragma unroll
    for (int c = 0; c < 16; ++c) {
        float acc = bias[c];
#pragma unroll
        for (int i = 0; i < 16; ++i) acc += patch[i] * s_w[c * 16 + i];
        z1[(b * 16 + c) * 16384 + oy * 128 + ox] = acc;
        atomicAdd(&s_sum[c], acc);
        atomicAdd(&s_sq[c], acc * acc);
    }
    __syncthreads();
    if (t < 16) {
        atomicAdd(&stats[ST_BN1_SUM + t], s_sum[t]);
        atomicAdd(&stats[ST_BN1_SUM + 16 + t], s_sq[t]);
    }
}

// ---------------- conv2: bn1+relu on load, 16->4, k4 s2 p1 -> z2(64,4,64,64) + bn2 stats
__global__ void k3_conv2(const float* __restrict__ z1, const float* __restrict__ w2,
                         const float* __restrict__ b2, float* stats,
                         float* __restrict__ z2) {
    __shared__ float s_w[1024];          // (co=4, ci=16, 4,4)
    __shared__ float s_scale[16], s_shift[16];
    __shared__ float s_sum[4], s_sq[4];
    int t = threadIdx.x;
    for (int i = t; i < 1024; i += 256) s_w[i] = w2[i];
    if (t < 16) { s_scale[t] = stats[ST_BN1_SC + t]; s_shift[t] = stats[ST_BN1_SC + 16 + t]; }
    if (t < 4)  { s_sum[t] = 0.f; s_sq[t] = 0.f; }
    __syncthreads();

    int gid = blockIdx.x * 256 + t;      // over 64*64*64
    int ox = gid & 63, oy = (gid >> 6) & 63, b = gid >> 12;
    const float* zb = z1 + b * 16 * 16384;
    __builtin_prefetch(zb + (2 * oy) * 128 + 2 * ox, 0, 0);   // global_prefetch_b8

    float acc[4] = { b2[0], b2[1], b2[2], b2[3] };
    for (int ky = 0; ky < 4; ++ky) {
        int iy = 2 * oy - 1 + ky;
        if ((unsigned)iy >= 128u) continue;
        for (int kx = 0; kx < 4; ++kx) {
            int ix = 2 * ox - 1 + kx;
            if ((unsigned)ix >= 128u) continue;
#pragma unroll
            for (int ci = 0; ci < 16; ++ci) {
                float v = zb[ci * 16384 + iy * 128 + ix];
                v = fmaxf(v * s_scale[ci] + s_shift[ci], 0.f);
#pragma unroll
                for (int co = 0; co < 4; ++co)
                    acc[co] += v * s_w[(co * 16 + ci) * 16 + ky * 4 + kx];
            }
        }
    }
#pragma unroll
    for (int co = 0; co < 4; ++co) {
        z2[(b * 4 + co) * 4096 + oy * 64 + ox] = acc[co];
        atomicAdd(&s_sum[co], acc[co]);
        atomicAdd(&s_sq[co], acc[co] * acc[co]);
    }
    __syncthreads();
    if (t < 4) {
        atomicAdd(&stats[ST_BN2_SUM + t], s_sum[t]);
        atomicAdd(&stats[ST_BN2_SUM + 4 + t], s_sq[t]);
    }
}

// ---------------- enc3(1x1,4->64) via WMMA f32 16x16x4 + VQ + loss + dec1-as-table ----------------
// Each wave handles 16 consecutive spatial positions of one batch image.
// Codebook table staged to LDS via the Tensor Data Mover (TENSORcnt tracked).
__global__ void k5_enc3_vq(const float* __restrict__ z2, const float* __restrict__ w3,
                           const float* __restrict__ b3, const float* __restrict__ emb,
                           float* stats, float* __restrict__ d1out,
                           float* __restrict__ tokens_out) {
    __shared__ float s_tile[8][16][68];  // [wave][row][col], stride 68 avoids bank conflicts
    __shared__ float s_emb[192];         // 3 x 64, filled by TDM
    __shared__ float s_loss;
    int t = threadIdx.x;
    int wave = t >> 5, lane = t & 31;

    if (t == 0) {
        s_loss = 0.f;
        // TDM: 2D tensor, 3 rows x 64 cols of f32, contiguous -> s_emb
        tdm_load_2d_f32_to_lds(emb, (unsigned)(unsigned long long)(&s_emb[0]), 64u, 3u, 64u);
    }
    if (t < 32) __builtin_amdgcn_s_wait_tensorcnt(0);   // wave 0 drains its TDM op
    __syncthreads();                                    // publish s_emb to all waves

    int tile = blockIdx.x * 8 + wave;          // 16384 tiles total
    int b = tile >> 8;
    int pos0 = (tile & 255) << 4;

    // A operand: 16x4 f32 tile of z2 (bn2+relu applied on load)
    int m = lane & 15;
    int kbase = (lane >> 4) << 1;              // 0 for lanes 0-15, 2 for lanes 16-31
    const float* z2b = z2 + b * 4 * 4096;
    float sc0 = stats[ST_BN2_SC + kbase],     sh0 = stats[ST_BN2_SC + 4 + kbase];
    float sc1 = stats[ST_BN2_SC + kbase + 1], sh1 = stats[ST_BN2_SC + 4 + kbase + 1];
    float a0 = fmaxf(z2b[(kbase + 0) * 4096 + pos0 + m] * sc0 + sh0, 0.f);
    float a1 = fmaxf(z2b[(kbase + 1) * 4096 + pos0 + m] * sc1 + sh1, 0.f);
    v2f a = { a0, a1 };

#pragma unroll
    for (int nt = 0; nt < 4; ++nt) {
        int n = nt * 16 + m;                   // N = lane%16 in B/C layouts
        v2f bv = { w3[n * 4 + kbase], w3[n * 4 + kbase + 1] };
        float bias = b3[n];
        v8f c = { bias, bias, bias, bias, bias, bias, bias, bias };
        c = __builtin_amdgcn_wmma_f32_16x16x4_f32(false, a, false, bv,
                                                  (short)0, c, false, false);
        int rbase = (lane < 16) ? 0 : 8;       // C layout: vgpr r, lanes>=16 -> M = r+8
#pragma unroll
        for (int r = 0; r < 8; ++r)
            s_tile[wave][rbase + r][nt * 16 + m] = c[r];
    }
    __syncthreads();

    // VQ: lanes 0..15 each own one row of the 16x64 zf tile
    float myloss = 0.f;
    if (lane < 16) {
        float d0 = 0.f, dd1 = 0.f, d2 = 0.f;
        for (int c = 0; c < 64; ++c) {
            float z = s_tile[wave][lane][c];
            float t0 = z - s_emb[c];        d0  += t0 * t0;
            float t1 = z - s_emb[64 + c];   dd1 += t1 * t1;
            float t2 = z - s_emb[128 + c];  d2  += t2 * t2;
        }
        int k = 0; float dmin = d0;
        if (dd1 < dmin) { dmin = dd1; k = 1; }
        if (d2  < dmin) { dmin = d2;  k = 2; }
        myloss = dmin;
        int hw = pos0 + lane;
        tokens_out[b * 4096 + hw] = (float)k;
#pragma unroll
        for (int co = 0; co < 4; ++co)
            d1out[(b * 4 + co) * 4096 + hw] = stats[ST_PROJ + k * 4 + co];
        atomicAdd(&s_loss, myloss);
    }
    __syncthreads();
    if (t == 0) atomicAdd(&stats[ST_LOSS], s_loss);
}

// ---------------- dec2: ConvT 4->16, k4 s2 p1, d1(64,4,64,64) -> y2(64,16,128,128) + bn3 stats
__global__ void k6_dec2(const float* __restrict__ d1in, const float* __restrict__ w,
                        const float* __restrict__ bias, float* __restrict__ y2,
                        float* stats) {
    __shared__ float s_w[1024];          // (ci=4, co=16, 4,4)
    __shared__ float s_sum[16], s_sq[16];
    int t = threadIdx.x;
    for (int i = t; i < 1024; i += 256) s_w[i] = w[i];
    if (t < 16) { s_sum[t] = 0.f; s_sq[t] = 0.f; }
    __syncthreads();

    int gid = blockIdx.x * 256 + t;      // over 64*128*128
    int ox = gid & 127, oy = (gid >> 7) & 127, b = gid >> 14;
    const float* db = d1in + b * 4 * 4096;
    __builtin_prefetch(db + (oy >> 1) * 64 + (ox >> 1), 0, 0);   // global_prefetch_b8

    float acc[16];
#pragma unroll
    for (int co = 0; co < 16; ++co) acc[co] = bias[co];

    int py = (oy + 1) & 1, px = (ox + 1) & 1;
#pragma unroll
    for (int dy = 0; dy < 2; ++dy) {
        int ky = py + 2 * dy;
        int iy = (oy + 1 - ky) >> 1;
        if ((unsigned)iy >= 64u) continue;
#pragma unroll
        for (int dx = 0; dx < 2; ++dx) {
            int kx = px + 2 * dx;
            int ix = (ox + 1 - kx) >> 1;
            if ((unsigned)ix >= 64u) continue;
#pragma unroll
            for (int ci = 0; ci < 4; ++ci) {
                float v = db[ci * 4096 + iy * 64 + ix];
#pragma unroll
                for (int co = 0; co < 16; ++co)
                    acc[co] += v * s_w[(ci * 16 + co) * 16 + ky * 4 + kx];
            }
        }
    }
#pragma unroll
    for (int co = 0; co < 16; ++co) {
        y2[(b * 16 + co) * 16384 + oy * 128 + ox] = acc[co];
        atomicAdd(&s_sum[co], acc[co]);
        atomicAdd(&s_sq[co], acc[co] * acc[co]);
    }
    __syncthreads();
    if (t < 16) {
        atomicAdd(&stats[ST_BN3_SUM + t], s_sum[t]);
        atomicAdd(&stats[ST_BN3_SUM + 16 + t], s_sq[t]);
    }
}

// ---------------- dec3: bn3+relu on load, ConvT 16->1, tanh -> y(64,1,256,256)
__global__ void k8_dec3(const float* __restrict__ y2, const float* __restrict__ w,
                        const float* __restrict__ bias, const float* __restrict__ stats,
                        float* __restrict__ yout) {
    __shared__ float s_w[256];           // (ci=16, 4,4)
    __shared__ float s_scale[16], s_shift[16];
    int t = threadIdx.x;
    s_w[t] = w[t];
    if (t < 16) { s_scale[t] = stats[ST_BN3_SC + t]; s_shift[t] = stats[ST_BN3_SC + 16 + t]; }
    __syncthreads();

    int gid = blockIdx.x * 256 + t;      // over 64*256*256
    int ox = gid & 255, oy = (gid >> 8) & 255, b = gid >> 16;
    const float* yb = y2 + b * 16 * 16384;
    __builtin_prefetch(yb + (oy >> 1) * 128 + (ox >> 1), 0, 0);  // global_prefetch_b8

    float acc = bias[0];
    int py = (oy + 1) & 1, px = (ox + 1) & 1;
#pragma unroll
    for (int dy = 0; dy < 2; ++dy) {
        int ky = py + 2 * dy;
        int iy = (oy + 1 - ky) >> 1;
        if ((unsigned)iy >= 128u) continue;
#pragma unroll
        for (int dx = 0; dx < 2; ++dx) {
            int kx = px + 2 * dx;
            int ix = (ox + 1 - kx) >> 1;
            if ((unsigned)ix >= 128u) continue;
#pragma unroll
            for (int ci = 0; ci < 16; ++ci) {
                float v = yb[ci * 16384 + iy * 128 + ix];
                v = fmaxf(v * s_scale[ci] + s_shift[ci], 0.f);
                acc += v * s_w[ci * 16 + ky * 4 + kx];
            }
        }
    }
    yout[gid] = tanhf(acc);
}

__global__ void k9_loss(const float* __restrict__ stats, float* __restrict__ out) {
    if (threadIdx.x == 0)
        out[0] = 1.2f * stats[ST_LOSS] * (1.0f / 16777216.0f);  // (1+BETA)*mean
}

// ---------------- host launcher ----------------
extern "C" void kernel_launch(void* const* d_in, const int* in_sizes, int n_in,
                              void* d_out, int out_size, void* d_ws, size_t ws_size,
                              hipStream_t stream) {
    (void)in_sizes; (void)n_in; (void)out_size; (void)ws_size;
    const float* x      = (const float*)d_in[0];
    const float* enc_w1 = (const float*)d_in[1];
    const float* enc_b1 = (const float*)d_in[2];
    const float* bn1_g  = (const float*)d_in[3];
    const float* bn1_b  = (const float*)d_in[4];
    const float* enc_w2 = (const float*)d_in[5];
    const float* enc_b2 = (const float*)d_in[6];
    const float* bn2_g  = (const float*)d_in[7];
    const float* bn2_b  = (const float*)d_in[8];
    const float* enc_w3 = (const float*)d_in[9];
    const float* enc_b3 = (const float*)d_in[10];
    const float* emb    = (const float*)d_in[11];
    const float* dec_w1 = (const float*)d_in[12];
    const float* dec_b1 = (const float*)d_in[13];
    const float* dec_w2 = (const float*)d_in[14];
    const float* dec_b2 = (const float*)d_in[15];
    const float* bn3_g  = (const float*)d_in[16];
    const float* bn3_b  = (const float*)d_in[17];
    const float* dec_w3 = (const float*)d_in[18];
    const float* dec_b3 = (const float*)d_in[19];

    float* ws     = (float*)d_ws;
    float* z1     = ws + Z1_OFF;         // reused as y2 after conv2 consumes it
    float* z2     = ws + Z2_OFF;
    float* d1buf  = ws + D1_OFF;
    float* stats  = ws + ST_OFF;

    float* y_out      = (float*)d_out;                 // 64*1*256*256
    float* loss_out   = (float*)d_out + 4194304;       // scalar
    float* tokens_out = (float*)d_out + 4194305;       // 64*64*64

    k_zero_stats<<<1, 256, 0, stream>>>(stats);
    k_proj<<<1, 64, 0, stream>>>(emb, dec_w1, dec_b1, stats);

    k1_conv1<<<4096, 256, 0, stream>>>(x, enc_w1, enc_b1, z1, stats);
    k_finalize_bn<<<1, 32, 0, stream>>>(stats, bn1_g, bn1_b, ST_BN1_SUM, ST_BN1_SC,
                                        16, 1.0f / 1048576.0f);
    k3_conv2<<<1024, 256, 0, stream>>>(z1, enc_w2, enc_b2, stats, z2);
    k_finalize_bn<<<1, 32, 0, stream>>>(stats, bn2_g, bn2_b, ST_BN2_SUM, ST_BN2_SC,
                                        4, 1.0f / 262144.0f);

    k5_enc3_vq<<<2048, 256, 0, stream>>>(z2, enc_w3, enc_b3, emb, stats,
                                         d1buf, tokens_out);

    k6_dec2<<<4096, 256, 0, stream>>>(d1buf, dec_w2, dec_b2, z1 /*y2*/, stats);
    k_finalize_bn<<<1, 32, 0, stream>>>(stats, bn3_g, bn3_b, ST_BN3_SUM, ST_BN3_SC,
                                        16, 1.0f / 1048576.0f);
    k8_dec3<<<16384, 256, 0, stream>>>(z1 /*y2*/, dec_w3, dec_b3, stats, y_out);

    k9_loss<<<1, 32, 0, stream>>>(stats, loss_out);
}